// Propagator_48284022342100
// MI455X (gfx1250) — compile-verified
//
#include <hip/hip_runtime.h>
#include <cstdint>
#include <cstddef>

// ---------------- static problem config (matches reference) ----------------
#define NZg   400            // raw model size
#define NL    424            // reference padded size (NZ + 2*TP)
#define TPAD  12             // reference total pad each side
#define NFR   436            // framed storage: 2 + 27*16 + 2
#define NFF   (NFR * NFR)
#define NSHOT 2
#define NSTEP 100
#define NRECV 100
#define NT    27             // 16x16 tiles per dim (covers 432 >= 424)
#define DXc   5.0f
#define DTc   0.0008f
#define DT2c  (DTc * DTc)

typedef __attribute__((ext_vector_type(2))) float v2f;
typedef __attribute__((ext_vector_type(8))) float v8f;

// ---- compile-time banded stencil coefficients (folded with 1/DX^n) --------
constexpr float ccd2(int d) {
    int a = d < 0 ? -d : d;
    return a == 0 ? (-2.5f / 25.0f)
         : a == 1 ? ((4.0f / 3.0f) / 25.0f)
         : a == 2 ? ((-1.0f / 12.0f) / 25.0f)
         : 0.0f;
}
constexpr float ccd1(int d) {
    int a = d < 0 ? -d : d;
    float m = a == 1 ? ((8.0f / 12.0f) / 5.0f)
            : a == 2 ? ((-1.0f / 12.0f) / 5.0f)
            : 0.0f;
    return d < 0 ? -m : m;
}
// Per-(chunk,lane) WMMA operand LUTs. Identical formula serves as the A
// operand (z-direction, 16x20 banded) and the B operand (y-direction, 20x16).
struct CoefTabs { alignas(16) float t2[5 * 32 * 2]; alignas(16) float t1[5 * 32 * 2]; };
constexpr CoefTabs buildTabs() {
    CoefTabs t{};
    for (int k = 0; k < 5; ++k)
        for (int l = 0; l < 32; ++l) {
            int m = l & 15, hi = l >> 4;
            int ka0 = 4 * k + 2 * hi, ka1 = ka0 + 1;
            t.t2[(k * 32 + l) * 2 + 0] = ccd2(ka0 - m - 2);
            t.t2[(k * 32 + l) * 2 + 1] = ccd2(ka1 - m - 2);
            t.t1[(k * 32 + l) * 2 + 0] = ccd1(ka0 - m - 2);
            t.t1[(k * 32 + l) * 2 + 1] = ccd1(ka1 - m - 2);
        }
    return t;
}
__device__ __constant__ CoefTabs g_tabs = buildTabs();

#define WMMA_F32(ACC, A, B) \
    (ACC) = __builtin_amdgcn_wmma_f32_16x16x4_f32(false, (A), false, (B), (short)0, (ACC), false, false)

// ---------------------------------------------------------------------------
// Fused time step: one wave (32 lanes) per 16x16 tile per shot.
//   - async-copies three 20x20 halo tiles (wfc, phi_z, phi_y) into LDS
//   - six banded matmuls via chained v_wmma_f32_16x16x4_f32 (K=20 as 5xK=4)
//   - FMA-only epilogue using precomputed time-invariant PML factors
//   - block 0 additionally records receivers of the *previous* step from wfc
// ---------------------------------------------------------------------------
__global__ __launch_bounds__(32)
void wave_step(const float* __restrict__ wfc,  float* __restrict__ wfout,  // wfout holds wfp, overwritten pointwise
               const float* __restrict__ phzi, float* __restrict__ phzo,
               const float* __restrict__ phyi, float* __restrict__ phyo,
               const float4* __restrict__ pack, // {2-dt2*sp, 1-0.5dt*ss, 1/(1+0.5dt*ss), dt2*c2}
               const float*  __restrict__ cz,   // dt*c2*(sig_z - sig_y)
               const float*  __restrict__ dsg,  // 1 - dt*sig (1-D, shared by z/y)
               const float*  __restrict__ amp,  const int* __restrict__ sloc,
               const int*    __restrict__ rloc, float* __restrict__ out, int t)
{
    __shared__ __align__(16) float s_w [20][20];
    __shared__ __align__(16) float s_pz[20][20];
    __shared__ __align__(16) float s_py[20][20];

    const int lane = threadIdx.x;
    const int bid  = blockIdx.x;
    const int shot = bid / (NT * NT);
    const int tz   = (bid / NT) % NT;
    const int ty   = bid % NT;
    const int z0   = tz * 16, y0 = ty * 16;
    const size_t soff = (size_t)shot * NFF;

    // deferred receiver recording of the previous step (wfc == wfn(t-1))
    if (t > 0 && bid == 0) {
        for (int i = lane; i < NSHOT * NRECV; i += 32) {
            int sh = i / NRECV, r = i - sh * NRECV;
            int rz = rloc[(sh * NRECV + r) * 2 + 0] + TPAD + 2;
            int ry = rloc[(sh * NRECV + r) * 2 + 1] + TPAD + 2;
            out[(size_t)(t - 1) * (NSHOT * NRECV) + i] =
                wfc[(size_t)sh * NFF + (size_t)rz * NFR + ry];
        }
    }

    // CDNA5 async copy: 3 tiles x 20 rows x 5 B128 segments = 300 transfers
    {
        const float* g0 = wfc  + soff;
        const float* g1 = phzi + soff;
        const float* g2 = phyi + soff;
        for (int i = lane; i < 300; i += 32) {
            int which = i / 100;
            int rem   = i - which * 100;
            int r = rem / 5, seg = rem - r * 5;
            const float* gp = (which == 0 ? g0 : which == 1 ? g1 : g2)
                              + (size_t)(z0 + r) * NFR + (y0 + seg * 4);
            float* lp = (which == 0 ? &s_w[0][0] : which == 1 ? &s_pz[0][0] : &s_py[0][0])
                        + r * 20 + seg * 4;
            unsigned           laddr = (unsigned)(uintptr_t)lp;
            unsigned long long ga    = (unsigned long long)(uintptr_t)gp;
            asm volatile("global_load_async_to_lds_b128 %0, %1, off"
                         :: "v"(laddr), "v"(ga) : "memory");
        }
    }

    // coefficient operand LUT loads overlap with the async copies
    const v2f* __restrict__ T2 = reinterpret_cast<const v2f*>(g_tabs.t2);
    const v2f* __restrict__ T1 = reinterpret_cast<const v2f*>(g_tabs.t1);

    asm volatile("s_wait_asynccnt 0x0" ::: "memory");

    // --- WMMA banded-matmul derivatives (EXEC all-1s here; single wave) ---
    const int m  = lane & 15;
    const int hi = lane >> 4;

    v8f a_d2y = {}, a_gy = {}, a_py = {};
    v8f a_d2z = {}, a_gz = {}, a_pz = {};

#pragma unroll
    for (int k = 0; k < 5; ++k) {
        const int ka0 = 4 * k + 2 * hi;
        const int ka1 = ka0 + 1;
        v2f aw, ap, bw, bp;
        // A operands (16x4 chunk): lane = row M, K striped per ISA layout
        aw.x = s_w [m + 2][ka0];  aw.y = s_w [m + 2][ka1];
        ap.x = s_py[m + 2][ka0];  ap.y = s_py[m + 2][ka1];
        // B operands (4x16 chunk): lane = col N, K striped
        bw.x = s_w [ka0][m + 2];  bw.y = s_w [ka1][m + 2];
        bp.x = s_pz[ka0][m + 2];  bp.y = s_pz[ka1][m + 2];
        // banded coefficient chunks from compile-time LUT
        v2f b2 = T2[k * 32 + lane];
        v2f b1 = T1[k * 32 + lane];

        WMMA_F32(a_d2y, aw, b2);   // d2 along y of wfc
        WMMA_F32(a_gy,  aw, b1);   // d1 along y of wfc
        WMMA_F32(a_py,  ap, b1);   // d1 along y of phi_y
        WMMA_F32(a_d2z, b2, bw);   // d2 along z of wfc   (const A, data B)
        WMMA_F32(a_gz,  b1, bw);   // d1 along z of wfc
        WMMA_F32(a_pz,  b1, bp);   // d1 along z of phi_z
    }

    // --- FMA-only epilogue: leapfrog + PML, injection, masked stores ---
    const int   szs  = sloc[shot * 2 + 0] + TPAD;
    const int   sys  = sloc[shot * 2 + 1] + TPAD;
    const float ampv = amp[t * NSHOT + shot];

#pragma unroll
    for (int r = 0; r < 8; ++r) {
        const int M = r + hi * 8;
        const int z = z0 + M, y = y0 + m;
        if (z < NL && y < NL) {
            const float lap = a_d2z[r] + a_d2y[r];
            const float wc  = s_w [M + 2][m + 2];
            const float pzc = s_pz[M + 2][m + 2];
            const float pyc = s_py[M + 2][m + 2];
            const size_t idx = soff + (size_t)(z + 2) * NFR + (y + 2);
            const float  wp  = wfout[idx];         // old wfp, pointwise read-before-write
            const int    ci  = z * NL + y;
            const float4 p   = pack[ci];
            const float  czv = cz[ci];
            float wn = (p.x * wc - p.y * wp + p.w * lap
                        + DT2c * (a_pz[r] + a_py[r])) * p.z;
            if (z == szs && y == sys) wn += ampv * p.w;   // amp * dt2 * c2
            wfout[idx] = wn;
            phzo[idx] = dsg[z] * pzc - czv * a_gz[r];
            phyo[idx] = dsg[y] * pyc + czv * a_gy[r];
        }
    }
}

// final-step receiver gather
__global__ void record_final(const float* __restrict__ wf, const int* __restrict__ rloc,
                             float* __restrict__ out)
{
    int i = threadIdx.x + blockIdx.x * blockDim.x;
    if (i < NSHOT * NRECV) {
        int sh = i / NRECV, r = i - sh * NRECV;
        int rz = rloc[(sh * NRECV + r) * 2 + 0] + TPAD + 2;
        int ry = rloc[(sh * NRECV + r) * 2 + 1] + TPAD + 2;
        out[i] = wf[(size_t)sh * NFF + (size_t)rz * NFR + ry];
    }
}

// zero all 6 field buffers (frames stay permanently zero) + reset max accumulator
__global__ void zero_fields(float* __restrict__ base, size_t n, int* __restrict__ maxbits)
{
    size_t i = (size_t)blockIdx.x * blockDim.x + threadIdx.x;
    if (i < n) base[i] = 0.0f;
    if (i == 0) *maxbits = 0;
}

// c^2 (edge-replicated pad) + max velocity via positive-float bit-pattern max
__global__ void prep_c2(const float* __restrict__ vp, float* __restrict__ c2,
                        int* __restrict__ maxbits)
{
    __shared__ int smax;
    if (threadIdx.x == 0) smax = 0;
    __syncthreads();
    int i = blockIdx.x * blockDim.x + threadIdx.x;
    int local = 0;
    if (i < NL * NL) {
        int z = i / NL, y = i - (i / NL) * NL;
        int iz = min(max(z - TPAD, 0), NZg - 1);
        int iy = min(max(y - TPAD, 0), NZg - 1);
        float cv = 1500.0f + 1000.0f * vp[iz * NZg + iy];
        c2[i] = cv * cv;
        local = __float_as_int(cv);
    }
    atomicMax(&smax, local);
    __syncthreads();
    if (threadIdx.x == 0) atomicMax(maxbits, smax);
}

// PML sigma profile (identical for z and y: both padded dims are 424)
__global__ void make_sigma(float* __restrict__ sig, const int* __restrict__ maxbits)
{
    int i = threadIdx.x;
    if (i < NL) {
        float maxvel = __int_as_float(*maxbits);
        const float scale = 3.0f * logf(1000.0f) / (2.0f * DXc * 10.0f);
        float v = 0.0f;
        if (i <= 1)                       { float x = 0.9f;                     v = x * x * scale * maxvel; }
        else if (i < 12)                  { float x = (float)(11 - i) * 0.1f;   v = x * x * scale * maxvel; }
        else if (i >= 412 && i < 422)     { float x = (float)(i - 412) * 0.1f;  v = x * x * scale * maxvel; }
        else if (i >= 422)                { float x = 0.9f;                     v = x * x * scale * maxvel; }
        sig[i] = v;
    }
}

// time-invariant PML update factors, hoisted out of the 100-step hot loop
__global__ void prep_coef(const float* __restrict__ c2, const float* __restrict__ sig,
                          float4* __restrict__ pack, float* __restrict__ cz,
                          float* __restrict__ dsg)
{
    int i = blockIdx.x * blockDim.x + threadIdx.x;
    if (i < NL * NL) {
        int z = i / NL, y = i - (i / NL) * NL;
        float sgz = sig[z], sgy = sig[y];
        float ss = sgz + sgy, sp = sgz * sgy;
        float c2v = c2[i];
        float4 p;
        p.x = 2.0f - DT2c * sp;
        p.y = 1.0f - 0.5f * DTc * ss;
        p.z = 1.0f / (1.0f + 0.5f * DTc * ss);
        p.w = DT2c * c2v;
        pack[i] = p;
        cz[i] = DTc * c2v * (sgz - sgy);
    }
    if (i < NL) dsg[i] = 1.0f - DTc * sig[i];
}

extern "C" void kernel_launch(void* const* d_in, const int* in_sizes, int n_in,
                              void* d_out, int out_size, void* d_ws, size_t ws_size,
                              hipStream_t stream)
{
    const float* vp   = (const float*)d_in[0];
    const float* amp  = (const float*)d_in[1];
    const int*   sloc = (const int*)d_in[2];
    const int*   rloc = (const int*)d_in[3];
    float*       out  = (float*)d_out;

    const size_t FS = (size_t)NSHOT * NFF;     // one field buffer (framed)
    float* ws   = (float*)d_ws;
    float* wfA  = ws;
    float* wfB  = wfA + FS;
    float* pzA  = wfB + FS;
    float* pzB  = pzA + FS;
    float* pyA  = pzB + FS;
    float* pyB  = pyA + FS;
    float4* pack = (float4*)(pyB + FS);                    // NL*NL float4
    float* czb  = (float*)(pack + (size_t)NL * NL);        // NL*NL
    float* c2b  = czb + (size_t)NL * NL;                   // NL*NL (init scratch)
    float* sig  = c2b + (size_t)NL * NL;                   // 448
    float* dsg  = sig + 448;                               // 448
    int*   maxbits = (int*)(dsg + 448);

    const size_t ztotal = 6 * FS;
    zero_fields<<<(unsigned)((ztotal + 255) / 256), 256, 0, stream>>>(wfA, ztotal, maxbits);
    prep_c2<<<(NL * NL + 255) / 256, 256, 0, stream>>>(vp, c2b, maxbits);
    make_sigma<<<1, 448, 0, stream>>>(sig, maxbits);
    prep_coef<<<(NL * NL + 255) / 256, 256, 0, stream>>>(c2b, sig, pack, czb, dsg);

    float* cur = wfA; float* prv = wfB;        // wfn written over wfp (pointwise-safe)
    float* pzi = pzA; float* pzo = pzB;
    float* pyi = pyA; float* pyo = pyB;
    for (int t = 0; t < NSTEP; ++t) {
        wave_step<<<NSHOT * NT * NT, 32, 0, stream>>>(cur, prv, pzi, pzo, pyi, pyo,
                                                      pack, czb, dsg, amp, sloc, rloc, out, t);
        float* tmp;
        tmp = cur; cur = prv; prv = tmp;
        tmp = pzi; pzi = pzo; pzo = tmp;
        tmp = pyi; pyi = pyo; pyo = tmp;
    }
    record_final<<<1, 256, 0, stream>>>(cur, rloc, out + (size_t)(NSTEP - 1) * NSHOT * NRECV);
}